// Model_22265110462506
// MI455X (gfx1250) — compile-verified
//
#include <hip/hip_runtime.h>
#include <hip/hip_bf16.h>
#include <math.h>

// Problem geometry (fixed by the reference): B=4, S=4096, H=4096
#define HDIM   4096
#define HDIM4  1024                 // float4 per row
#define NROWS  16384                // B*S
#define NELEM  67108864LL           // B*S*H
#define TPB    256                  // 8 wave32s per block

// Native clang vector (required by __builtin_nontemporal_load/store;
// HIP's float4 is a struct and is rejected).
typedef float v4f __attribute__((ext_vector_type(4)));

// ---------------------------------------------------------------------------
// Kernel 0: zero the global abs-max accumulator (ws is poisoned by harness)
// ---------------------------------------------------------------------------
__global__ void init_absmax(unsigned* g_absmax) {
    if (threadIdx.x == 0) *g_absmax = 0u;
}

// ---------------------------------------------------------------------------
// Pass 1: x = x1 + x2 (written out), per-row inv-RMS, global abs-max of
//         (x * invrms * gamma).  One row (4096 f32) per block.
// x1/x2 are read-once -> non-temporal loads (TH_NT) so they don't evict the
// x writeback lines from the 192 MB L2; x itself is stored RT so pass 2 can
// hit in L2 for a large fraction of its re-read.
// ---------------------------------------------------------------------------
__global__ __launch_bounds__(TPB) void pass1_rms_absmax(
    const v4f* __restrict__ x1, const v4f* __restrict__ x2,
    const v4f* __restrict__ gamma,
    v4f* __restrict__ x_out, float* __restrict__ row_inv,
    unsigned* __restrict__ g_absmax)
{
    const int  r    = blockIdx.x;
    const int  t    = threadIdx.x;
    const long base = (long)r * HDIM4;

    v4f   s[4];
    float ss = 0.0f;
#pragma unroll
    for (int j = 0; j < 4; ++j) {
        const int c = t + j * TPB;           // float4 column, fully coalesced
        v4f a = __builtin_nontemporal_load(&x1[base + c]);
        v4f b = __builtin_nontemporal_load(&x2[base + c]);
        v4f v = a + b;
        s[j] = v;
        x_out[base + c] = v;                 // residual output, RT (L2-resident)
        ss += v.x * v.x + v.y * v.y + v.z * v.z + v.w * v.w;
    }

    // wave32 reduction, then cross-wave via LDS
#pragma unroll
    for (int o = 16; o > 0; o >>= 1) ss += __shfl_xor(ss, o, 32);

    __shared__ float red[TPB / 32];
    __shared__ float s_inv;
    const int wave = t >> 5, lane = t & 31;
    if (lane == 0) red[wave] = ss;
    __syncthreads();
    if (t == 0) {
        float tot = 0.0f;
#pragma unroll
        for (int w = 0; w < TPB / 32; ++w) tot += red[w];
        const float rms = sqrtf(tot * (1.0f / (float)HDIM) + 1e-6f);
        const float inv = 1.0f / rms;
        s_inv = inv;
        row_inv[r] = inv;
    }
    __syncthreads();
    const float inv = s_inv;

    float am = 0.0f;
#pragma unroll
    for (int j = 0; j < 4; ++j) {
        const int c = t + j * TPB;
        v4f g = gamma[c];
        v4f v = s[j];
        am = fmaxf(am, fabsf(v.x * inv * g.x));
        am = fmaxf(am, fabsf(v.y * inv * g.y));
        am = fmaxf(am, fabsf(v.z * inv * g.z));
        am = fmaxf(am, fabsf(v.w * inv * g.w));
    }
#pragma unroll
    for (int o = 16; o > 0; o >>= 1) am = fmaxf(am, __shfl_xor(am, o, 32));
    if (lane == 0) red[wave] = am;           // WAR ordered by barrier above
    __syncthreads();
    if (t == 0) {
        float m = 0.0f;
#pragma unroll
        for (int w = 0; w < TPB / 32; ++w) m = fmaxf(m, red[w]);
        // all values >= 0 -> IEEE bit pattern is monotonic under uint compare
        atomicMax(g_absmax, __float_as_uint(m));
    }
}

// ---------------------------------------------------------------------------
// Pass 2: stream x back through LDS via CDNA5 async global->LDS engine,
//         recompute out = x * invrms * gamma, int8-quantize with the global
//         scale, write y1 and y2 (as float, per harness output dtype).
// Wave-local LDS slots: each lane async-loads exactly the bytes it will read
// back, so `s_wait_asynccnt 0` alone gives consistency (no barrier needed).
// y1/y2 are write-once -> non-temporal stores (TH_NT), no L2 residency.
// ---------------------------------------------------------------------------
__global__ __launch_bounds__(TPB) void pass2_quant(
    const float* __restrict__ x, const v4f* __restrict__ gamma,
    const float* __restrict__ row_inv, const unsigned* __restrict__ g_absmax,
    v4f* __restrict__ y1, v4f* __restrict__ y2,
    float* __restrict__ scales)
{
    const int  r    = blockIdx.x;
    const int  t    = threadIdx.x;
    const long base = (long)r * HDIM;        // float units

    __shared__ float tile[HDIM];             // 16 KB of the WGP's 320 KB LDS

    // Kick off 4 async b128 copies per lane: HBM/L2 -> LDS (ASYNCcnt tracked)
#pragma unroll
    for (int j = 0; j < 4; ++j) {
        const int      c   = (t + j * TPB) * 4;                 // float index
        const unsigned lds = (unsigned)(unsigned long long)(&tile[c]);
        const float*   ga  = x + base + c;
        asm volatile("global_load_async_to_lds_b128 %0, %1, off"
                     :: "v"(lds), "v"(ga) : "memory");
    }

    // Overlap: fetch scalars while the async engine streams the row
    const float maxv  = __uint_as_float(*g_absmax);
    const float scale = maxv * (1.0f / 127.0f);   // max/127
    const float rcp   = 127.0f / maxv;            // == 1/scale
    const float inv   = row_inv[r];

    asm volatile("s_wait_asynccnt 0x0" ::: "memory");

    const long b4 = (long)r * HDIM4;
#pragma unroll
    for (int j = 0; j < 4; ++j) {
        const int c4 = t + j * TPB;
        v4f v = *(const v4f*)&tile[c4 * 4];   // ds_load_b128
        v4f g = gamma[c4];
        v4f q;
        q.x = fminf(fmaxf(rintf(v.x * inv * g.x * rcp), -128.0f), 127.0f);
        q.y = fminf(fmaxf(rintf(v.y * inv * g.y * rcp), -128.0f), 127.0f);
        q.z = fminf(fmaxf(rintf(v.z * inv * g.z * rcp), -128.0f), 127.0f);
        q.w = fminf(fmaxf(rintf(v.w * inv * g.w * rcp), -128.0f), 127.0f);
        __builtin_nontemporal_store(q, &y1[b4 + c4]);
        __builtin_nontemporal_store(q, &y2[b4 + c4]);
    }

    if (r == 0 && t == 0) {
        scales[0] = scale;   // scale1
        scales[1] = scale;   // scale2 (identical in the reference)
    }
}

// ---------------------------------------------------------------------------
// Launch: d_in = {x1, x2, gamma, smooth_scale1(unused), smooth_scale2(unused)}
// d_out (float) = [ y1 (N) | y2 (N) | x (N) | scale1 | scale2 ]
// ---------------------------------------------------------------------------
extern "C" void kernel_launch(void* const* d_in, const int* in_sizes, int n_in,
                              void* d_out, int out_size, void* d_ws, size_t ws_size,
                              hipStream_t stream)
{
    const float* x1    = (const float*)d_in[0];
    const float* x2    = (const float*)d_in[1];
    const float* gamma = (const float*)d_in[2];

    float* out    = (float*)d_out;
    float* y1     = out;                 // N floats
    float* y2     = out + NELEM;         // N floats
    float* x_res  = out + 2 * NELEM;     // N floats
    float* scales = out + 3 * NELEM;     // 2 floats

    unsigned* g_absmax = (unsigned*)d_ws;
    float*    row_inv  = (float*)((char*)d_ws + 256);   // NROWS floats

    init_absmax<<<1, 1, 0, stream>>>(g_absmax);

    pass1_rms_absmax<<<NROWS, TPB, 0, stream>>>(
        (const v4f*)x1, (const v4f*)x2, (const v4f*)gamma,
        (v4f*)x_res, row_inv, g_absmax);

    pass2_quant<<<NROWS, TPB, 0, stream>>>(
        x_res, (const v4f*)gamma, row_inv, g_absmax,
        (v4f*)y1, (v4f*)y2, scales);
}